// HAGERecConv_16844861735263
// MI455X (gfx1250) — compile-verified
//
#include <hip/hip_runtime.h>
#include <math.h>

typedef __attribute__((ext_vector_type(2))) float v2f;
typedef __attribute__((ext_vector_type(8))) float v8f;

#define D_FEAT 64
#define NHEAD 2
#define AFEAT 32
#define NEG_SLOPE 0.2f

// ---------------------------------------------------------------------------
// float atomic max via sign-split int/uint trick (lowers to
// global_atomic_max_i32 / global_atomic_min_u32)
__device__ __forceinline__ void atomicMaxF(float* addr, float v) {
    if (v >= 0.0f) {
        atomicMax((int*)addr, __float_as_int(v));
    } else {
        atomicMin((unsigned int*)addr, __float_as_uint(v));
    }
}

// ---------------------------------------------------------------------------
// Kernel 1: en = feat @ w2s + b2s ; ev = feat @ w2d + b2d
// block = 256 threads = 8 waves. wave w: sel = w/4 (en vs ev), coltile = w%4.
// All 16 A/B fragments preloaded (load clause), then 16 back-to-back WMMAs.
__global__ void node_gemm_scores(const float* __restrict__ feat,
                                 const float* __restrict__ w2s, const float* __restrict__ b2s,
                                 const float* __restrict__ w2d, const float* __restrict__ b2d,
                                 float* __restrict__ en, float* __restrict__ ev, int n) {
    const int row0 = blockIdx.x * 16;
    const int wave = threadIdx.x >> 5;
    const int lane = threadIdx.x & 31;
    const int sel  = wave >> 2;        // 0 -> en/w2s, 1 -> ev/w2d
    const int ct   = wave & 3;         // column tile (16 cols each)
    const float* W = sel ? w2d : w2s;
    const float* B = sel ? b2d : b2s;
    float* OUT     = sel ? ev  : en;

    const int m   = lane & 15;               // row within tile (A) / col (B,C)
    const int kk  = (lane >> 4) << 1;        // 0 or 2: K sub-offset per half-wave
    const int col0 = ct * 16;
    int row = row0 + m;
    int rowc = row < n ? row : (n - 1);      // clamp: no EXEC divergence on loads

    // Preload A fragments: lane reads half the feat row as 8-byte chunks.
    const float* ap = feat + (size_t)rowc * D_FEAT + kk;
    v2f afr[16];
    #pragma unroll
    for (int i = 0; i < 16; ++i)
        afr[i] = *(const v2f*)(ap + 4 * i);

    // Preload B fragments (column-strided scalar loads).
    const float* wp = W + (size_t)kk * D_FEAT + col0 + m;
    v2f bfr[16];
    #pragma unroll
    for (int i = 0; i < 16; ++i) {
        bfr[i].x = wp[256 * i];
        bfr[i].y = wp[256 * i + 64];
    }

    v8f c = {};
    #pragma unroll
    for (int i = 0; i < 16; ++i)
        c = __builtin_amdgcn_wmma_f32_16x16x4_f32(false, afr[i], false, bfr[i],
                                                  (short)0, c, false, false);

    const float bias = B[col0 + m];
    const int outrow0 = row0 + ((lane >> 4) << 3);   // lanes 16-31 hold rows 8..15
    #pragma unroll
    for (int v = 0; v < 8; ++v) {
        int r = outrow0 + v;
        if (r < n) OUT[(size_t)r * D_FEAT + col0 + m] = c[v] + bias;
    }
}

// ---------------------------------------------------------------------------
// Kernel 2: init workspace (must run every call: determinism)
__global__ void init_ws(float* __restrict__ seg_max, float* __restrict__ seg_sum,
                        float* __restrict__ g_sum, int nH, int nD) {
    int i = blockIdx.x * blockDim.x + threadIdx.x;
    if (i < nH) { seg_max[i] = -INFINITY; seg_sum[i] = 0.0f; }
    if (i < nD) g_sum[i] = 0.0f;
}

// ---------------------------------------------------------------------------
// Kernel 3: per-(edge,head) attention logit + segment max (atomic)
__global__ void edge_scores(const float* __restrict__ en, const float* __restrict__ ev,
                            const float* __restrict__ attn,
                            const int* __restrict__ src, const int* __restrict__ dst,
                            float* __restrict__ sc, float* __restrict__ seg_max, int e_cnt) {
    int t = blockIdx.x * blockDim.x + threadIdx.x;
    if (t >= e_cnt * NHEAD) return;
    int e = t >> 1, h = t & 1;
    int s = src[e], d = dst[e];
    const float* pe = en + (size_t)s * D_FEAT + h * AFEAT;
    const float* pv = ev + (size_t)d * D_FEAT + h * AFEAT;
    const float* pa = attn + h * AFEAT;
    float acc = 0.0f;
    #pragma unroll 8
    for (int i = 0; i < AFEAT; ++i) {
        float x = pe[i] + pv[i];
        x = (x >= 0.0f) ? x : NEG_SLOPE * x;
        acc = fmaf(x, pa[i], acc);
    }
    sc[(size_t)e * NHEAD + h] = acc;
    atomicMaxF(&seg_max[(size_t)d * NHEAD + h], acc);
}

// ---------------------------------------------------------------------------
// Kernel 4: ex = exp(s - max[dst]) (in-place), segment sum (atomic)
__global__ void edge_exp(const int* __restrict__ dst, float* __restrict__ sc,
                         const float* __restrict__ seg_max, float* __restrict__ seg_sum,
                         int e_cnt) {
    int t = blockIdx.x * blockDim.x + threadIdx.x;
    if (t >= e_cnt * NHEAD) return;
    int e = t >> 1, h = t & 1;
    int d = dst[e];
    float m = seg_max[(size_t)d * NHEAD + h];
    float ex = expf(sc[(size_t)e * NHEAD + h] - m);
    sc[(size_t)e * NHEAD + h] = ex;
    atomicAdd(&seg_sum[(size_t)d * NHEAD + h], ex);
}

// ---------------------------------------------------------------------------
// Kernel 5: one wave32 per edge. msg = softmax_D(feat[src]*r_feat*alpha),
// heads summed in-register, atomic scatter into g_sum[dst].
__global__ void edge_message(const float* __restrict__ feat, const float* __restrict__ r_feat,
                             const int* __restrict__ src, const int* __restrict__ dst,
                             const float* __restrict__ sc, const float* __restrict__ seg_sum,
                             float* __restrict__ g_sum, int e_cnt) {
    int wid  = (blockIdx.x * blockDim.x + threadIdx.x) >> 5;
    int lane = threadIdx.x & 31;
    if (wid >= e_cnt) return;
    int s = src[wid], d = dst[wid];

    float f0 = feat[(size_t)s * D_FEAT + lane];
    float f1 = feat[(size_t)s * D_FEAT + 32 + lane];
    float r0 = r_feat[(size_t)wid * D_FEAT + lane];
    float r1 = r_feat[(size_t)wid * D_FEAT + 32 + lane];

    float acc0 = 0.0f, acc1 = 0.0f;
    #pragma unroll
    for (int h = 0; h < NHEAD; ++h) {
        float a = sc[(size_t)wid * NHEAD + h] / seg_sum[(size_t)d * NHEAD + h];
        float x0 = f0 * r0 * a;
        float x1 = f1 * r1 * a;
        // wave-wide max over 64 elements (2 per lane)
        float m = fmaxf(x0, x1);
        for (int off = 16; off > 0; off >>= 1)
            m = fmaxf(m, __shfl_xor(m, off, 32));
        float e0 = expf(x0 - m);
        float e1 = expf(x1 - m);
        float ssum = e0 + e1;
        for (int off = 16; off > 0; off >>= 1)
            ssum += __shfl_xor(ssum, off, 32);
        float inv = 1.0f / ssum;
        acc0 = fmaf(e0, inv, acc0);
        acc1 = fmaf(e1, inv, acc1);
    }
    atomicAdd(&g_sum[(size_t)d * D_FEAT + lane], acc0);
    atomicAdd(&g_sum[(size_t)d * D_FEAT + 32 + lane], acc1);
}

// ---------------------------------------------------------------------------
// Kernel 6: out = (feat + g) @ w1 + b1 + (feat * g) @ w4 + b4
// block = 128 threads = 4 waves; wave w owns column tile w. K is processed in
// two chunks of 8 steps: preload fragment arrays (one load clause), then 16
// back-to-back WMMAs sharing one accumulator.
__global__ void final_gemm(const float* __restrict__ feat, const float* __restrict__ g_sum,
                           const float* __restrict__ w1, const float* __restrict__ b1,
                           const float* __restrict__ w4, const float* __restrict__ b4,
                           float* __restrict__ out, int n) {
    const int row0 = blockIdx.x * 16;
    const int wave = threadIdx.x >> 5;   // 0..3 -> column tile
    const int lane = threadIdx.x & 31;
    const int m   = lane & 15;
    const int kk  = (lane >> 4) << 1;
    const int col0 = wave * 16;
    int row = row0 + m;
    int rowc = row < n ? row : (n - 1);  // clamp: no EXEC divergence on loads

    const float* fp  = feat  + (size_t)rowc * D_FEAT + kk;
    const float* gp  = g_sum + (size_t)rowc * D_FEAT + kk;
    const float* w1p = w1 + (size_t)kk * D_FEAT + col0 + m;
    const float* w4p = w4 + (size_t)kk * D_FEAT + col0 + m;

    v8f c = {};
    #pragma unroll
    for (int chunk = 0; chunk < 2; ++chunk) {
        v2f fv[8], gv[8], b1v[8], b4v[8];
        #pragma unroll
        for (int i = 0; i < 8; ++i) {
            int ks = chunk * 8 + i;
            fv[i] = *(const v2f*)(fp + 4 * ks);
            gv[i] = *(const v2f*)(gp + 4 * ks);
            b1v[i].x = w1p[256 * ks];
            b1v[i].y = w1p[256 * ks + 64];
            b4v[i].x = w4p[256 * ks];
            b4v[i].y = w4p[256 * ks + 64];
        }
        #pragma unroll
        for (int i = 0; i < 8; ++i) {
            v2f a1 = fv[i] + gv[i];   // (feat + g)
            v2f a2 = fv[i] * gv[i];   // (feat * g)
            c = __builtin_amdgcn_wmma_f32_16x16x4_f32(false, a1, false, b1v[i],
                                                      (short)0, c, false, false);
            c = __builtin_amdgcn_wmma_f32_16x16x4_f32(false, a2, false, b4v[i],
                                                      (short)0, c, false, false);
        }
    }

    const float bias = b1[col0 + m] + b4[col0 + m];
    const int outrow0 = row0 + ((lane >> 4) << 3);
    #pragma unroll
    for (int v = 0; v < 8; ++v) {
        int r = outrow0 + v;
        if (r < n) out[(size_t)r * D_FEAT + col0 + m] = c[v] + bias;
    }
}

// ---------------------------------------------------------------------------
extern "C" void kernel_launch(void* const* d_in, const int* in_sizes, int n_in,
                              void* d_out, int out_size, void* d_ws, size_t ws_size,
                              hipStream_t stream) {
    const float* feat   = (const float*)d_in[0];
    const float* r_feat = (const float*)d_in[1];
    const float* w1_w   = (const float*)d_in[2];
    const float* w1_b   = (const float*)d_in[3];
    const float* w2s_w  = (const float*)d_in[4];
    const float* w2s_b  = (const float*)d_in[5];
    const float* w2d_w  = (const float*)d_in[6];
    const float* w2d_b  = (const float*)d_in[7];
    const float* w4_w   = (const float*)d_in[8];
    const float* w4_b   = (const float*)d_in[9];
    const float* attn   = (const float*)d_in[10];
    const int*   src    = (const int*)d_in[11];
    const int*   dst    = (const int*)d_in[12];

    const int N = in_sizes[0] / D_FEAT;
    const int E = in_sizes[11];

    // workspace layout (floats)
    float* ws      = (float*)d_ws;
    float* en      = ws;                               // N*64
    float* ev      = en + (size_t)N * D_FEAT;          // N*64
    float* sc      = ev + (size_t)N * D_FEAT;          // E*2
    float* seg_max = sc + (size_t)E * NHEAD;           // N*2
    float* seg_sum = seg_max + (size_t)N * NHEAD;      // N*2
    float* g_sum   = seg_sum + (size_t)N * NHEAD;      // N*64
    float* out     = (float*)d_out;

    const int node_tiles = (N + 15) / 16;

    // 1) en/ev GEMMs (WMMA)
    node_gemm_scores<<<node_tiles, 256, 0, stream>>>(feat, w2s_w, w2s_b, w2d_w, w2d_b,
                                                     en, ev, N);
    // 2) init segment buffers + g accumulator
    {
        int total = N * D_FEAT;   // >= N*NHEAD
        init_ws<<<(total + 255) / 256, 256, 0, stream>>>(seg_max, seg_sum, g_sum,
                                                         N * NHEAD, total);
    }
    // 3) edge logits + segment max
    edge_scores<<<(E * NHEAD + 255) / 256, 256, 0, stream>>>(en, ev, attn, src, dst,
                                                             sc, seg_max, E);
    // 4) exp + segment sum
    edge_exp<<<(E * NHEAD + 255) / 256, 256, 0, stream>>>(dst, sc, seg_max, seg_sum, E);
    // 5) per-edge message softmax + scatter (wave32 per edge, 8 waves/block)
    edge_message<<<(E + 7) / 8, 256, 0, stream>>>(feat, r_feat, src, dst,
                                                  sc, seg_sum, g_sum, E);
    // 6) fused final GEMMs (WMMA)
    final_gemm<<<node_tiles, 128, 0, stream>>>(feat, g_sum, w1_w, w1_b, w4_w, w4_b,
                                               out, N);
}